// SoftDTW_63737314673164
// MI455X (gfx1250) — compile-verified
//
#include <hip/hip_runtime.h>
#include <hip/hip_bf16.h>

#define BATCH 32
#define NSEQ  512
#define MSEQ  512
#define DIM   128
#define INF_F 1.0e10f

typedef float v2f __attribute__((ext_vector_type(2)));
typedef float v8f __attribute__((ext_vector_type(8)));

// ---------------------------------------------------------------------------
// Phase 0: row norms  nx[b,n] = sum_k x[b,n,k]^2  (and same for y)
// ---------------------------------------------------------------------------
__global__ void row_norms_kernel(const float* __restrict__ x,
                                 const float* __restrict__ y,
                                 float* __restrict__ nx,
                                 float* __restrict__ ny) {
  int r = blockIdx.x * blockDim.x + threadIdx.x;
  if (r >= BATCH * NSEQ) return;
  const float* p = x + (size_t)r * DIM;
  const float* q = y + (size_t)r * DIM;
  float sx = 0.f, sy = 0.f;
#pragma unroll 4
  for (int k = 0; k < DIM; ++k) {
    sx = fmaf(p[k], p[k], sx);
    sy = fmaf(q[k], q[k], sy);
  }
  nx[r] = sx;
  ny[r] = sy;
}

// ---------------------------------------------------------------------------
// Phase 1: D = nx + ny - 2 * (x @ y^T) via V_WMMA_F32_16X16X4_F32.
// One wave computes a 16x16 tile of D for one batch; K=128 in steps of 4.
// A fragment (16x4 f32, 2 VGPRs): lane L holds row m=L%16,
//   VGPR0/1 = K = k0 + 2*(L/16) + {0,1}  -> one float2 load.
// B = y^T, so B[k][n] = y[n][k]: same float2 indexing on y's rows.
// ---------------------------------------------------------------------------
__global__ void __launch_bounds__(128)
sqdist_wmma_kernel(const float* __restrict__ x,
                   const float* __restrict__ y,
                   const float* __restrict__ nx,
                   const float* __restrict__ ny,
                   float* __restrict__ D) {
  const int lane = threadIdx.x & 31;
  const int wave = threadIdx.x >> 5;
  const int tile = blockIdx.x * 4 + wave;   // 1024 tiles per batch
  const int tm   = tile >> 5;               // 0..31  (rows of D)
  const int tn   = tile & 31;               // 0..31  (cols of D)
  const int b    = blockIdx.y;

  const float* xb = x + (size_t)b * NSEQ * DIM;
  const float* yb = y + (size_t)b * MSEQ * DIM;

  const int lr = lane & 15;        // row within 16-wide fragment
  const int kh = (lane >> 4) * 2;  // K sub-offset for upper half-wave

  const float* xrow = xb + (size_t)(tm * 16 + lr) * DIM + kh;
  const float* yrow = yb + (size_t)(tn * 16 + lr) * DIM + kh;

  v8f acc = {};
#pragma unroll
  for (int k = 0; k < DIM; k += 4) {
    v2f a  = *(const v2f*)(xrow + k);
    v2f bf = *(const v2f*)(yrow + k);
    // (neg_a, A, neg_b, B, c_mod, C, reuse_a, reuse_b)
    acc = __builtin_amdgcn_wmma_f32_16x16x4_f32(false, a, false, bf,
                                                (short)0, acc, false, false);
  }

  // C/D layout: element v of acc -> (m = tm*16 + v + (lane<16?0:8), n = tn*16 + lane%16)
  const float yn  = ny[b * MSEQ + tn * 16 + lr];
  const int   mb  = tm * 16 + ((lane < 16) ? 0 : 8);
  const float* nxp = nx + b * NSEQ + mb;
  float* Dp = D + ((size_t)b * NSEQ + mb) * MSEQ + tn * 16 + lr;
#pragma unroll
  for (int v = 0; v < 8; ++v) {
    Dp[(size_t)v * MSEQ] = nxp[v] + yn - 2.0f * acc[v];
  }
}

// ---------------------------------------------------------------------------
// Phase 2: anti-diagonal soft-DTW DP. One workgroup (512 threads = 16 waves)
// per batch; thread i owns row i, j = d - i on diagonal d. Three rotating
// 512-float rows in LDS, one barrier per diagonal. gamma = 1.
// D lives in the 192MB L2, each thread streams its row sequentially.
// ---------------------------------------------------------------------------
__global__ void __launch_bounds__(NSEQ)
softdtw_dp_kernel(const float* __restrict__ D, float* __restrict__ partial) {
  const int b = blockIdx.x;
  const int i = threadIdx.x;
  __shared__ float R[3][NSEQ];
  R[0][i] = INF_F;
  R[1][i] = INF_F;
  __syncthreads();

  const float* Drow = D + ((size_t)b * NSEQ + i) * MSEQ;
  float last = INF_F;
  for (int d = 0; d < NSEQ + MSEQ - 1; ++d) {
    const int p2 = d % 3, p1 = (d + 1) % 3, cur = (d + 2) % 3;
    const int j = d - i;
    // r0 = R[i-1, j-1], r1 = R[i-1, j], r2 = R[i, j-1]
    float r0 = (i == 0) ? ((d == 0) ? 0.0f : INF_F) : R[p2][i - 1];
    float r1 = (i == 0) ? INF_F : R[p1][i - 1];
    float r2 = R[p1][i];
    float mn = fminf(r0, fminf(r1, r2));
    float s  = __expf(mn - r0) + __expf(mn - r1) + __expf(mn - r2);
    float softmin = mn - __logf(s);            // gamma = 1
    float val = INF_F;
    if (j >= 0 && j < MSEQ) {
      val = Drow[j] + softmin;
      if (j + 32 < MSEQ) __builtin_prefetch(Drow + j + 32, 0, 0);
    }
    R[cur][i] = val;
    last = val;
    __syncthreads();
  }
  // result = R[d = N+M-2][i = N-1] (j = M-1), i.e. thread 511's last val
  if (i == NSEQ - 1) partial[b] = last;
}

// ---------------------------------------------------------------------------
// Phase 3: mean over 32 batch results
// ---------------------------------------------------------------------------
__global__ void reduce_mean_kernel(const float* __restrict__ partial,
                                   float* __restrict__ out) {
  if (threadIdx.x == 0) {
    float t = 0.f;
    for (int i = 0; i < BATCH; ++i) t += partial[i];
    out[0] = t / (float)BATCH;
  }
}

extern "C" void kernel_launch(void* const* d_in, const int* in_sizes, int n_in,
                              void* d_out, int out_size, void* d_ws, size_t ws_size,
                              hipStream_t stream) {
  const float* x = (const float*)d_in[0];
  const float* y = (const float*)d_in[1];
  float* out = (float*)d_out;

  // workspace layout
  float* D       = (float*)d_ws;                       // 32*512*512 floats (33.5 MB)
  float* nx      = D + (size_t)BATCH * NSEQ * MSEQ;    // 32*512
  float* ny      = nx + BATCH * NSEQ;                  // 32*512
  float* partial = ny + BATCH * MSEQ;                  // 32

  row_norms_kernel<<<(BATCH * NSEQ + 255) / 256, 256, 0, stream>>>(x, y, nx, ny);
  sqdist_wmma_kernel<<<dim3(256, BATCH), 128, 0, stream>>>(x, y, nx, ny, D);
  softdtw_dp_kernel<<<BATCH, NSEQ, 0, stream>>>(D, partial);
  reduce_mean_kernel<<<1, 32, 0, stream>>>(partial, out);
}